// RandomShifts_20968030339608
// MI455X (gfx1250) — compile-verified
//
#include <hip/hip_runtime.h>

// RandomShifts: out[n,c,i,j] = x[n,c, clamp(i+sy-4,0,83), clamp(j+sx-4,0,83)]
// Pure bandwidth op (~260 MB total @ 23.3 TB/s -> ~11 us). Strategy:
//   - stage each (n,c) 84x84 plane into LDS via async 128-bit global->LDS DMA
//   - clamped gather from LDS, 128-bit non-temporal stores to global.

#define PADV 4
#define HH 84
#define WW 84
#define CC 9
#define NN 512
#define PLANE (HH * WW)            // 7056 floats = 28224 B
#define V4_PER_PLANE (PLANE / 4)   // 1764
#define THREADS 256

typedef float v4f __attribute__((ext_vector_type(4)));
typedef int   v4i __attribute__((ext_vector_type(4)));

// AS-qualified pointee typedefs (AS1 = global, AS3 = LDS)
typedef __attribute__((address_space(1))) v4i as1_v4i;
typedef __attribute__((address_space(3))) v4i as3_v4i;

#if defined(__has_builtin)
#if __has_builtin(__builtin_amdgcn_global_load_async_to_lds_b128)
#define HAVE_ASYNC_LDS 1
#endif
#if __has_builtin(__builtin_amdgcn_s_wait_asynccnt)
#define HAVE_WAIT_ASYNC 1
#endif
#endif
#ifndef HAVE_ASYNC_LDS
#define HAVE_ASYNC_LDS 0
#endif
#ifndef HAVE_WAIT_ASYNC
#define HAVE_WAIT_ASYNC 0
#endif

__device__ __forceinline__ void g2lds_b128(const float* g, float* l) {
#if HAVE_ASYNC_LDS
  __builtin_amdgcn_global_load_async_to_lds_b128(
      (as1_v4i*)g, (as3_v4i*)l, /*offset=*/0, /*cpol=*/0);
#else
  *(v4f*)l = *(const v4f*)g;
#endif
}

__device__ __forceinline__ void wait_async_lds() {
#if HAVE_ASYNC_LDS
#if HAVE_WAIT_ASYNC
  __builtin_amdgcn_s_wait_asynccnt(0);
#else
  asm volatile("s_wait_asynccnt 0" ::: "memory");
#endif
#endif
}

__global__ __launch_bounds__(THREADS) void RandomShifts_20968030339608_kernel(
    const float* __restrict__ x, const int* __restrict__ shift,
    float* __restrict__ out) {
  __shared__ __attribute__((aligned(16))) float tile[PLANE];

  const int plane = blockIdx.x;        // plane = n*CC + c  (uniform)
  const int n = plane / CC;
  const int t = threadIdx.x;
  const float* __restrict__ src = x + (size_t)plane * PLANE;
  float* __restrict__ dst = out + (size_t)plane * PLANE;

  // ---- Stage the full source plane into LDS: 1764 x 16B async transfers ----
#pragma unroll
  for (int k = 0; k < (V4_PER_PLANE + THREADS - 1) / THREADS; ++k) {
    int v = t + k * THREADS;
    if (v < V4_PER_PLANE) g2lds_b128(src + v * 4, tile + v * 4);
  }
  wait_async_lds();   // each wave drains its own ASYNCcnt
  __syncthreads();    // cross-wave visibility of LDS writes

  // Uniform per-block shift -> scalar loads
  const int sx = shift[2 * n + 0] - PADV;  // dx in [-4, 4]
  const int sy = shift[2 * n + 1] - PADV;  // dy in [-4, 4]

  // ---- Clamped gather from LDS, 128-bit NT stores ----
#pragma unroll
  for (int k = 0; k < (V4_PER_PLANE + THREADS - 1) / THREADS; ++k) {
    int v = t + k * THREADS;
    if (v < V4_PER_PLANE) {
      int i = v / (WW / 4);            // output row
      int j0 = (v % (WW / 4)) * 4;     // first output column of this vec4
      int sr = i + sy;
      sr = sr < 0 ? 0 : (sr > HH - 1 ? HH - 1 : sr);
      const float* row = tile + sr * WW;
      v4f o;
      int j;
      j = j0 + 0 + sx; j = j < 0 ? 0 : (j > WW - 1 ? WW - 1 : j); o.x = row[j];
      j = j0 + 1 + sx; j = j < 0 ? 0 : (j > WW - 1 ? WW - 1 : j); o.y = row[j];
      j = j0 + 2 + sx; j = j < 0 ? 0 : (j > WW - 1 ? WW - 1 : j); o.z = row[j];
      j = j0 + 3 + sx; j = j < 0 ? 0 : (j > WW - 1 ? WW - 1 : j); o.w = row[j];
      __builtin_nontemporal_store(o, (v4f*)(dst + v * 4));
    }
  }
}

extern "C" void kernel_launch(void* const* d_in, const int* in_sizes, int n_in,
                              void* d_out, int out_size, void* d_ws, size_t ws_size,
                              hipStream_t stream) {
  (void)in_sizes; (void)n_in; (void)out_size; (void)d_ws; (void)ws_size;
  const float* x = (const float*)d_in[0];
  const int* shift = (const int*)d_in[1];
  float* out = (float*)d_out;
  dim3 grid(NN * CC);   // 4608 planes
  dim3 block(THREADS);
  RandomShifts_20968030339608_kernel<<<grid, block, 0, stream>>>(x, shift, out);
}